// keops_RBFkernel_2705829396576
// MI455X (gfx1250) — compile-verified
//
#include <hip/hip_runtime.h>

typedef __attribute__((ext_vector_type(2))) float v2f;
typedef __attribute__((ext_vector_type(4))) float v4f;
typedef __attribute__((ext_vector_type(8))) float v8f;

#define NROWS 32768
#define MCOLS 32768
#define DDIM  32
#define TILE  16
#define SLICES 4
#define WAVES_PER_BLOCK 8
#define ROWTILES_PER_BLOCK (WAVES_PER_BLOCK / SLICES)   // 2
#define ROWS_PER_BLOCK (ROWTILES_PER_BLOCK * TILE)      // 32
#define JTILES (MCOLS / TILE)                           // 2048
#define TILE_FLOATS (TILE * DDIM)                       // 512 floats / tile

__device__ __forceinline__ v2f vlo(v4f v) { v2f r; r.x = v.x; r.y = v.y; return r; }
__device__ __forceinline__ v2f vhi(v4f v) { v2f r; r.x = v.z; r.y = v.w; return r; }

// ---------------------------------------------------------------------------
// Prep 1: ws[0..N) = c*||x_i||^2 ; ws[N..N+M) = c*||y_j||^2 ; c = -ls*log2(e)
// ---------------------------------------------------------------------------
__global__ __launch_bounds__(256) void rbf_prep_sq(const float* __restrict__ ls,
                                                   const float* __restrict__ x,
                                                   const float* __restrict__ y,
                                                   float* __restrict__ ws) {
  int i = blockIdx.x * blockDim.x + threadIdx.x;
  if (i >= NROWS + MCOLS) return;
  float c = -ls[0] * 1.44269504088896340736f;
  const float* src = (i < NROWS) ? (x + (size_t)i * DDIM)
                                 : (y + (size_t)(i - NROWS) * DDIM);
  float s = 0.f;
#pragma unroll
  for (int k = 0; k < DDIM; ++k) s = __builtin_fmaf(src[k], src[k], s);
  ws[i] = c * s;
}

// ---------------------------------------------------------------------------
// Prep 2: repack y into WMMA B-operand order so each lane's per-tile data is
// 64 contiguous bytes:  pack[tile][lane][kc][0..1] = y[tile*16 + (lane&15)]
//                                                     [4*kc + 2*(lane>>4) + 0..1]
// ---------------------------------------------------------------------------
__global__ __launch_bounds__(256) void rbf_prep_pack(const float* __restrict__ y,
                                                     float* __restrict__ pack) {
  int idx = blockIdx.x * blockDim.x + threadIdx.x;   // tile*32 + lane
  if (idx >= JTILES * 32) return;
  int tile = idx >> 5;
  int lane = idx & 31;
  int l15  = lane & 15;
  int koff = (lane >> 4) * 2;
  const float* yrow = y + (size_t)(tile * TILE + l15) * DDIM + koff;
  float* dst = pack + (size_t)idx * 16;
#pragma unroll
  for (int kc = 0; kc < 8; ++kc) {
    dst[2 * kc + 0] = yrow[4 * kc + 0];
    dst[2 * kc + 1] = yrow[4 * kc + 1];
  }
}

// ---------------------------------------------------------------------------
// Main: one wave = one 16-row x tile x one j-slice.
// 8 chained V_WMMA_F32_16X16X4_F32 per tile; e = cx + cy - 2c*S; acc += 2^e.
// ---------------------------------------------------------------------------
__global__ __launch_bounds__(256) void rbf_main(const float* __restrict__ ls,
                                                const float* __restrict__ x,
                                                const float* __restrict__ pack,
                                                const float* __restrict__ cx,
                                                const float* __restrict__ wy,
                                                float* __restrict__ out) {
  __shared__ float red[SLICES][ROWS_PER_BLOCK];

  const int tid   = threadIdx.x;
  const int wave  = tid >> 5;
  const int lane  = tid & 31;
  const int slice = wave & (SLICES - 1);
  const int rtloc = wave >> 2;                        // 0..1
  const int mbase = (blockIdx.x * ROWTILES_PER_BLOCK + rtloc) * TILE;

  const int l15  = lane & 15;
  const int half = lane >> 4;     // 0: K=0,1 ; 1: K=2,3  (A/B VGPR layout)
  const int koff = half * 2;

  const float c   = -ls[0] * 1.44269504088896340736f;
  const float m2c = -2.0f * c;    // coefficient of S inside the exponent

  // --- A matrix (x tile), resident for the whole sweep: 8 chunks of K=4 ---
  const float* xrow = x + (size_t)(mbase + l15) * DDIM + koff;
  v2f a[8];
#pragma unroll
  for (int kc = 0; kc < 8; ++kc) a[kc] = *(const v2f*)(xrow + 4 * kc);

  // --- c*||x||^2 for the 8 rows this lane accumulates (C-matrix layout) ---
  float cxv[8];
#pragma unroll
  for (int v = 0; v < 8; ++v) cxv[v] = cx[mbase + v + 8 * half];

  const int jtiles = JTILES / SLICES;                 // 512 tiles per slice
  const int nt0    = slice * jtiles;

  const float* pbase = pack + (size_t)nt0 * TILE_FLOATS + lane * 16;
  const float* wyp   = wy + nt0 * TILE + l15;

  v8f acc = {0.f, 0.f, 0.f, 0.f, 0.f, 0.f, 0.f, 0.f};

#pragma unroll 2
  for (int t = 0; t < jtiles; ++t) {
    const float* pt = pbase + (size_t)t * TILE_FLOATS;
    __builtin_prefetch(pt + 8 * TILE_FLOATS, 0, 1);   // stream-ahead into L0

    float cyq = wyp[t * TILE];                        // c*||y_n||^2, n = l15

    // 64 contiguous bytes per lane -> 4x global_load_b128 in one clause
    v4f q0 = *(const v4f*)(pt + 0);
    v4f q1 = *(const v4f*)(pt + 4);
    v4f q2 = *(const v4f*)(pt + 8);
    v4f q3 = *(const v4f*)(pt + 12);

    v2f b[8];
    b[0] = vlo(q0); b[1] = vhi(q0);
    b[2] = vlo(q1); b[3] = vhi(q1);
    b[4] = vlo(q2); b[5] = vhi(q2);
    b[6] = vlo(q3); b[7] = vhi(q3);

    v8f s = {0.f, 0.f, 0.f, 0.f, 0.f, 0.f, 0.f, 0.f};
#pragma unroll
    for (int kc = 0; kc < 8; ++kc) {
      s = __builtin_amdgcn_wmma_f32_16x16x4_f32(
          false, a[kc], false, b[kc], (short)0, s, false, false);
    }

#pragma unroll
    for (int v = 0; v < 8; ++v) {
      float e = __builtin_fmaf(m2c, s[v], cxv[v] + cyq);
      acc[v] += __builtin_amdgcn_exp2f(e);            // v_exp_f32
    }
  }

  // --- reduce over the 16 columns: butterfly inside each 16-lane half ---
#pragma unroll
  for (int m = 1; m < 16; m <<= 1) {
#pragma unroll
    for (int v = 0; v < 8; ++v) acc[v] += __shfl_xor(acc[v], m, 32);
  }

  // lane 0 holds rows mbase+0..7, lane 16 holds rows mbase+8..15
  if (lane == 0 || lane == 16) {
#pragma unroll
    for (int v = 0; v < 8; ++v)
      red[slice][rtloc * TILE + 8 * half + v] = acc[v];
  }
  __syncthreads();

  // --- deterministic cross-slice combine, one thread per output row ---
  if (tid < ROWS_PER_BLOCK) {
    float s = 0.f;
#pragma unroll
    for (int sl = 0; sl < SLICES; ++sl) s += red[sl][tid];
    out[blockIdx.x * ROWS_PER_BLOCK + tid] = s;
  }
}

// ---------------------------------------------------------------------------
extern "C" void kernel_launch(void* const* d_in, const int* in_sizes, int n_in,
                              void* d_out, int out_size, void* d_ws, size_t ws_size,
                              hipStream_t stream) {
  const float* ls = (const float*)d_in[0];
  const float* x  = (const float*)d_in[1];
  const float* y  = (const float*)d_in[2];
  float* out  = (float*)d_out;
  float* ws   = (float*)d_ws;           // needs (N + M + M*32)*4 ~= 4.5 MB
  float* cx   = ws;
  float* wy   = ws + NROWS;
  float* pack = ws + NROWS + MCOLS;

  rbf_prep_sq<<<(NROWS + MCOLS + 255) / 256, 256, 0, stream>>>(ls, x, y, ws);
  rbf_prep_pack<<<(JTILES * 32 + 255) / 256, 256, 0, stream>>>(y, pack);
  rbf_main<<<NROWS / ROWS_PER_BLOCK, 256, 0, stream>>>(ls, x, pack, cx, wy, out);
}